// ColumnwiseCosineSimilarityLoss_7035156431183
// MI455X (gfx1250) — compile-verified
//
#include <hip/hip_runtime.h>
#include <math.h>

// Columnwise cosine similarity loss for [16384, 4096] fp32 inputs.
// HBM-bound: 512 MB read -> ~22us floor at 23.3 TB/s.
// Row reduction runs through V_WMMA_F32_16X16X4_F32 with an all-ones A matrix
// (D = 1*B + C  =>  per-column sum over K), which is exact fp32 accumulation.

typedef float v2f __attribute__((ext_vector_type(2)));
typedef float v8f __attribute__((ext_vector_type(8)));

#define NCOLS 4096
#define NROWS 16384
#define ROWCHUNKS 32
#define ROWS_PER_CHUNK (NROWS / ROWCHUNKS)   // 512
#define COLGROUPS (NCOLS / 128)              // 32 (8 waves x 16 cols per block)

__global__ __launch_bounds__(256) void cosloss_stage1(
    const float* __restrict__ O, const float* __restrict__ T,
    float* __restrict__ pdot, float* __restrict__ pon, float* __restrict__ ptn)
{
    const int lane = threadIdx.x & 31;
    const int wave = threadIdx.x >> 5;
    // wave handles a 16-column tile; lane L owns column (L % 16)
    const int col  = blockIdx.x * 128 + wave * 16 + (lane & 15);
    const int half = lane >> 4;            // which row of the B-register pair
    const int rowBase = blockIdx.y * ROWS_PER_CHUNK;

    v2f ones;
    ones.x = 1.0f;
    ones.y = 1.0f;                         // A = all-ones 16x4 (layout-invariant)

    v8f acc_ot = {};                       // sum o*t per column (replicated over M)
    v8f acc_oo = {};                       // sum o*o
    v8f acc_tt = {};                       // sum t*t

    const float* op = O + (size_t)(rowBase + half) * NCOLS + col;
    const float* tp = T + (size_t)(rowBase + half) * NCOLS + col;

    // 4 rows per iteration: rows {r+half, r+2+half} fill the two B VGPRs.
    // Each global load: half-wave covers 64B contiguous; the block's 8 waves
    // cover 128 adjacent columns, so cachelines are fully consumed.
    #pragma unroll 4
    for (int r = 0; r < ROWS_PER_CHUNK; r += 4) {
        float o0 = op[0];
        float o1 = op[2 * NCOLS];
        float t0 = tp[0];
        float t1 = tp[2 * NCOLS];
        op += 4 * NCOLS;
        tp += 4 * NCOLS;

        v2f b_ot, b_oo, b_tt;
        b_ot.x = o0 * t0;  b_ot.y = o1 * t1;
        b_oo.x = o0 * o0;  b_oo.y = o1 * o1;
        b_tt.x = t0 * t0;  b_tt.y = t1 * t1;

        // D = ones(16x4) x B(4x16) + C : reduces K, result per column n=lane%16
        acc_ot = __builtin_amdgcn_wmma_f32_16x16x4_f32(
            false, ones, false, b_ot, (short)0, acc_ot, false, false);
        acc_oo = __builtin_amdgcn_wmma_f32_16x16x4_f32(
            false, ones, false, b_oo, (short)0, acc_oo, false, false);
        acc_tt = __builtin_amdgcn_wmma_f32_16x16x4_f32(
            false, ones, false, b_tt, (short)0, acc_tt, false, false);
    }

    // Every acc element m holds the identical column sum (A was all ones);
    // lanes 0..15 and 16..31 duplicate columns, so emit from lanes 0..15 only.
    if (lane < 16) {
        size_t idx = (size_t)blockIdx.y * NCOLS + col;
        pdot[idx] = acc_ot[0];
        pon[idx]  = acc_oo[0];
        ptn[idx]  = acc_tt[0];
    }
}

__global__ __launch_bounds__(256) void cosloss_stage2(
    const float* __restrict__ pdot, const float* __restrict__ pon,
    const float* __restrict__ ptn, float* __restrict__ bsum)
{
    __shared__ float s[256];
    const int c = blockIdx.x * 256 + threadIdx.x;

    float dot = 0.0f, on2 = 0.0f, tn2 = 0.0f;
    #pragma unroll 8
    for (int rc = 0; rc < ROWCHUNKS; ++rc) {
        size_t idx = (size_t)rc * NCOLS + c;
        dot += pdot[idx];
        on2 += pon[idx];
        tn2 += ptn[idx];
    }
    const float EPS = 1e-8f;
    float cosv = dot / (fmaxf(sqrtf(on2), EPS) * fmaxf(sqrtf(tn2), EPS));

    s[threadIdx.x] = cosv;
    __syncthreads();
    #pragma unroll
    for (int off = 128; off > 0; off >>= 1) {
        if (threadIdx.x < off) s[threadIdx.x] += s[threadIdx.x + off];
        __syncthreads();
    }
    if (threadIdx.x == 0) bsum[blockIdx.x] = s[0];
}

__global__ void cosloss_stage3(const float* __restrict__ bsum,
                               float* __restrict__ out)
{
    if (threadIdx.x == 0) {
        float s = 0.0f;
        #pragma unroll
        for (int i = 0; i < NCOLS / 256; ++i) s += bsum[i];
        out[0] = 1.0f - s / (float)NCOLS;
    }
}

extern "C" void kernel_launch(void* const* d_in, const int* in_sizes, int n_in,
                              void* d_out, int out_size, void* d_ws, size_t ws_size,
                              hipStream_t stream) {
    const float* O = (const float*)d_in[0];
    const float* T = (const float*)d_in[1];

    float* ws   = (float*)d_ws;
    float* pdot = ws;
    float* pon  = ws + (size_t)ROWCHUNKS * NCOLS;
    float* ptn  = ws + (size_t)2 * ROWCHUNKS * NCOLS;
    float* bsum = ws + (size_t)3 * ROWCHUNKS * NCOLS;   // 16 floats

    dim3 g1(COLGROUPS, ROWCHUNKS);                      // 32 x 32 = 1024 blocks
    cosloss_stage1<<<g1, 256, 0, stream>>>(O, T, pdot, pon, ptn);
    cosloss_stage2<<<NCOLS / 256, 256, 0, stream>>>(pdot, pon, ptn, bsum);
    cosloss_stage3<<<1, 32, 0, stream>>>(bsum, (float*)d_out);
}